// DeformableDETRLoss_17514876634195
// MI455X (gfx1250) — compile-verified
//
#include <hip/hip_runtime.h>
#include <hip/hip_bf16.h>

#define B_  1024
#define Q_  900
#define C_  8
#define NT_ 32

typedef __attribute__((ext_vector_type(2))) float v2f;
typedef __attribute__((ext_vector_type(8))) float v8f;

// ---------------------------------------------------------------------------
// Wave32 sum using V_WMMA_F32_16X16X4_F32 as a cross-lane reducer.
// A (16x4): VGPR0 = v (A[m][0]=v[m], A[m][2]=v[m+16]), VGPR1 = 0.
// B (4x16): all ones.  => D[m][n] = v[m] + v[m+16] (f32 exact adds).
// Lanes 0-15: d[j] = v[j]+v[j+16]  -> sum d = v[0..7]+v[16..23]
// Lanes 16-31: d[j] = v[j+8]+v[j+24] -> sum d = v[8..15]+v[24..31]
// One shfl_xor(16) completes the 32-lane total on every lane.
// Requires EXEC == all ones (callers guarantee full-wave execution).
// ---------------------------------------------------------------------------
__device__ __forceinline__ float wave_sum32(float v) {
    v2f a; a[0] = v;    a[1] = 0.0f;
    v2f b; b[0] = 1.0f; b[1] = 1.0f;
    v8f c = {};
    v8f d = __builtin_amdgcn_wmma_f32_16x16x4_f32(
        /*neg_a=*/false, a, /*neg_b=*/false, b,
        /*c_mod=*/(short)0, c, /*reuse_a=*/false, /*reuse_b=*/false);
    float s = d[0] + d[1] + d[2] + d[3] + d[4] + d[5] + d[6] + d[7];
    s += __shfl_xor(s, 16, 32);
    return s;
}

// Block-level sum (blockDim.x multiple of 32, <= 256 here). Result valid on
// wave 0; thread 0 uses it. Ends with a barrier so LDS can be reused.
__device__ __forceinline__ float block_sum(float v, float* lds, int nwaves) {
    float w = wave_sum32(v);
    int wid  = threadIdx.x >> 5;
    int lane = threadIdx.x & 31;
    if (lane == 0) lds[wid] = w;
    __syncthreads();
    float r = 0.0f;
    if (wid == 0) {
        float x = (lane < nwaves) ? lds[lane] : 0.0f;
        r = wave_sum32(x);
    }
    __syncthreads();
    return r;
}

// ---------------------------------------------------------------------------
// Kernel 1: init workspace (target map = background class, zero accumulators)
// ---------------------------------------------------------------------------
__global__ void __launch_bounds__(256) ddetr_init_kernel(
    unsigned char* __restrict__ target, int* __restrict__ card,
    float* __restrict__ acc, int n) {
    int i = blockIdx.x * 256 + threadIdx.x;
    if (i < n)   target[i] = (unsigned char)C_;
    if (i < B_)  card[i] = 0;
    if (i < 4)   acc[i] = 0.0f;
}

// ---------------------------------------------------------------------------
// Kernel 2: scatter matched labels into the target-class map
// ---------------------------------------------------------------------------
__global__ void __launch_bounds__(256) ddetr_scatter_kernel(
    const int* __restrict__ sidx, const int* __restrict__ tlab,
    unsigned char* __restrict__ target) {
    int i = blockIdx.x * 256 + threadIdx.x;    // (b, n), n < 32
    int b = i >> 5;
    int q = sidx[i];
    target[b * Q_ + q] = (unsigned char)tlab[i];
}

// ---------------------------------------------------------------------------
// Kernel 3: sigmoid focal CE + cardinality.  One thread per (b,q) row.
// Streams 29.5 MB of logits via two aligned float4 loads per row.
// ---------------------------------------------------------------------------
__global__ void __launch_bounds__(256) ddetr_focal_card_kernel(
    const float* __restrict__ logits, const unsigned char* __restrict__ target,
    const float* __restrict__ ew, float* __restrict__ acc,
    int* __restrict__ card) {
    __shared__ float lds[8];
    int i = blockIdx.x * 256 + threadIdx.x;    // flat (b,q); grid is exact
    int b = i / Q_;
    int t = (int)target[i];
    float w = ew[t];

    const float4* rp = (const float4*)(logits + (size_t)i * 8);
    float4 lo = rp[0];
    float4 hi = rp[1];
    float xs[8] = {lo.x, lo.y, lo.z, lo.w, hi.x, hi.y, hi.z, hi.w};

    float rowsum = 0.0f;
    float mx = xs[0];
#pragma unroll
    for (int c = 0; c < 8; ++c) {
        float x  = xs[c];
        mx = fmaxf(mx, x);
        bool on  = (c == t);
        float ax = fabsf(x);
        float e  = __expf(-ax);                       // v_exp_f32
        float bce = fmaxf(x, 0.0f) - (on ? x : 0.0f) + __logf(1.0f + e);
        float inv = __builtin_amdgcn_rcpf(1.0f + e);  // v_rcp_f32
        float p   = (x >= 0.0f) ? inv : (1.0f - inv); // sigmoid(x)
        float pt  = on ? p : (1.0f - p);
        float omp = 1.0f - pt;
        rowsum += (on ? 0.25f : 0.75f) * omp * omp * bce;
    }
    float ce = rowsum * w;

    // cardinality: max sigmoid > 0.5  <=>  max logit > 0
    int pred = (mx > 0.0f) ? 1 : 0;
    unsigned long long m = __ballot(pred);
    int ibase = blockIdx.x * 256 + (threadIdx.x & ~31);
    int bf = ibase / Q_;
    int bl = (ibase + 31) / Q_;
    int lane = threadIdx.x & 31;
    if (bf == bl) {                 // wave-uniform batch index: 1 atomic/wave
        if (lane == 0) atomicAdd(&card[b], (int)__popcll(m));
    } else {                        // rare boundary wave
        if (pred) atomicAdd(&card[b], 1);
    }

    float bs = block_sum(ce, lds, 8);
    if (threadIdx.x == 0) atomicAdd(&acc[0], bs);
}

// ---------------------------------------------------------------------------
// Kernel 4: matched-box L1 + paired GIoU.  One thread per (b,n) match.
// ---------------------------------------------------------------------------
__global__ void __launch_bounds__(256) ddetr_box_kernel(
    const float* __restrict__ pboxes, const float* __restrict__ tboxes,
    const int* __restrict__ sidx, const int* __restrict__ tlab,
    float* __restrict__ acc) {
    __shared__ float lds[8];
    int i = blockIdx.x * 256 + threadIdx.x;    // (b, n); grid exact
    int b = i >> 5;
    int q = sidx[i];
    int lab = tlab[i];

    float4 s = *(const float4*)(pboxes + ((size_t)b * Q_ + q) * 4);
    float4 t = *(const float4*)(tboxes + (size_t)i * 4);

    float scale = (lab == 4 || lab == 5 || lab == 6) ? 2.0f : 1.0f;
    float l1 = fabsf(s.x - t.x) + fabsf(s.y - t.y) +
               fabsf(s.z - t.z) + fabsf(s.w - t.w);

    // cxcywh -> xyxy
    float ax0 = s.x - 0.5f * s.z, ay0 = s.y - 0.5f * s.w;
    float ax1 = s.x + 0.5f * s.z, ay1 = s.y + 0.5f * s.w;
    float bx0 = t.x - 0.5f * t.z, by0 = t.y - 0.5f * t.w;
    float bx1 = t.x + 0.5f * t.z, by1 = t.y + 0.5f * t.w;

    float areaA = (ax1 - ax0) * (ay1 - ay0);
    float areaB = (bx1 - bx0) * (by1 - by0);
    float iw = fmaxf(fminf(ax1, bx1) - fmaxf(ax0, bx0), 0.0f);
    float ih = fmaxf(fminf(ay1, by1) - fmaxf(ay0, by0), 0.0f);
    float inter = iw * ih;
    float uni = areaA + areaB - inter;
    float iou = inter / uni;
    float ew_ = fmaxf(fmaxf(ax1, bx1) - fminf(ax0, bx0), 0.0f);
    float eh_ = fmaxf(fmaxf(ay1, by1) - fminf(ay0, by0), 0.0f);
    float areaE = ew_ * eh_;
    float giou = iou - (areaE - uni) / areaE;

    float lb = l1 * scale;
    float lg = (1.0f - giou) * scale;

    float sb = block_sum(lb, lds, 8);
    float sg = block_sum(lg, lds, 8);
    if (threadIdx.x == 0) {
        atomicAdd(&acc[1], sb);
        atomicAdd(&acc[2], sg);
    }
}

// ---------------------------------------------------------------------------
// Kernel 5: cardinality mean + final weighting.  1 block of 1024 threads.
// ---------------------------------------------------------------------------
__global__ void __launch_bounds__(1024) ddetr_finalize_kernel(
    const float* __restrict__ acc, const int* __restrict__ card,
    float* __restrict__ out) {
    __shared__ float lds[32];
    int tid  = threadIdx.x;
    int wid  = tid >> 5;
    int lane = tid & 31;

    float v = fabsf((float)card[tid] - (float)NT_);
    float w = wave_sum32(v);
    if (lane == 0) lds[wid] = w;
    __syncthreads();
    if (wid == 0) {
        float total = wave_sum32(lds[lane]);
        if (lane == 0) {
            const float inv_nb = 1.0f / (float)(B_ * NT_);  // +1e-8 < 1 ulp
            out[0] = 1.0f * acc[0] * inv_nb;                // W_CE
            out[1] = 5.0f * acc[1] * inv_nb;                // W_BBOX
            out[2] = 2.0f * acc[2] * inv_nb;                // W_GIOU
            out[3] = 1.0f * (total * (1.0f / (float)B_));   // W_CARD
        }
    }
}

// ---------------------------------------------------------------------------
extern "C" void kernel_launch(void* const* d_in, const int* in_sizes, int n_in,
                              void* d_out, int out_size, void* d_ws, size_t ws_size,
                              hipStream_t stream) {
    const float* logits = (const float*)d_in[0];   // [1024,900,8] f32
    const float* pboxes = (const float*)d_in[1];   // [1024,900,4] f32
    const float* tboxes = (const float*)d_in[2];   // [1024,32,4]  f32
    const int*   sidx   = (const int*)d_in[3];     // [1024,32]    i32
    const int*   tlab   = (const int*)d_in[4];     // [1024,32]    i32
    const float* ew     = (const float*)d_in[5];   // [9]          f32
    float* out = (float*)d_out;

    char* ws = (char*)d_ws;
    float*         acc    = (float*)ws;                        // 4 floats
    int*           card   = (int*)(ws + 16);                   // 1024 ints
    unsigned char* target = (unsigned char*)(ws + 16 + 4096);  // B*Q bytes

    const int BQ  = B_ * Q_;     // 921600 = 3600 * 256 (exact)
    const int BNT = B_ * NT_;    // 32768  = 128  * 256 (exact)

    ddetr_init_kernel   <<<(BQ + 255) / 256, 256, 0, stream>>>(target, card, acc, BQ);
    ddetr_scatter_kernel<<<BNT / 256,        256, 0, stream>>>(sidx, tlab, target);
    ddetr_focal_card_kernel<<<BQ / 256,      256, 0, stream>>>(logits, target, ew, acc, card);
    ddetr_box_kernel    <<<BNT / 256,        256, 0, stream>>>(pboxes, tboxes, sidx, tlab, acc);
    ddetr_finalize_kernel<<<1,              1024, 0, stream>>>(acc, card, out);
}